// HetDGCN_7584912245183
// MI455X (gfx1250) — compile-verified
//
#include <hip/hip_runtime.h>
#include <cstdint>
#include <cstddef>

#define ZEROF 1e-8f
#define NN    8192      // N = B*NPG
#define DD    256       // D
#define BBLK  32        // B
#define NPGC  256       // NPG
#define EE    131072    // E
#define LLAY  2

typedef __attribute__((ext_vector_type(2))) float v2f;
typedef __attribute__((ext_vector_type(8))) float v8f;

__device__ __forceinline__ float fast_rcp(float x) { return __builtin_amdgcn_rcpf(x); }

// ---------------------------------------------------------------------------
// row norms: one wave (32 lanes) per row of H (N x 256)
// ---------------------------------------------------------------------------
__global__ void rownorm_kernel(const float* __restrict__ H, float* __restrict__ nrm) {
  const int row  = blockIdx.x * 8 + (threadIdx.x >> 5);
  const int lane = threadIdx.x & 31;
  const float* p = H + (size_t)row * DD + lane;
  float s = 0.f;
#pragma unroll
  for (int k = 0; k < DD; k += 32) { float v = p[k]; s += v * v; }
#pragma unroll
  for (int m = 16; m > 0; m >>= 1) s += __shfl_xor(s, m, 32);
  if (lane == 0) nrm[row] = sqrtf(s);
}

// ---------------------------------------------------------------------------
// S = P * Q^T  (P,Q: NN x 256 rowmajor), K = 256.
// One wave computes a 2x2 group of 16x16 tiles (32x32 output) so each
// fragment load feeds two v_wmma_f32_16x16x4_f32 ops (4 loads : 4 wmma).
// mode 0 (CORR):     A[i,j] = S * rcp(nrm_i*nrm_j + eps)
// mode 1 (SIGBLEND): A[i,j] = 0.5*A[i,j] + 0.5*sigmoid(S)
// ---------------------------------------------------------------------------
__global__ void gemm_nt_kernel(const float* __restrict__ P, const float* __restrict__ Q,
                               const float* __restrict__ nrm, float* __restrict__ A,
                               int mode) {
  const int wave = threadIdx.x >> 5;
  const int lane = threadIdx.x & 31;
  const int grp  = blockIdx.x * 8 + wave;        // 256*256 groups of 32x32
  const int gi = grp >> 8;
  const int gj = grp & 255;
  const int row0 = gi * 32, col0 = gj * 32;
  const int half = lane >> 4, idx = lane & 15;
  const int kb = half * 2;                       // K sub-offset per lane half
  const float* pr0 = P + (size_t)(row0 + idx) * DD + kb;
  const float* pr1 = pr0 + (size_t)16 * DD;
  const float* qr0 = Q + (size_t)(col0 + idx) * DD + kb;
  const float* qr1 = qr0 + (size_t)16 * DD;
  v8f c00 = {0.f,0.f,0.f,0.f,0.f,0.f,0.f,0.f};
  v8f c01 = c00, c10 = c00, c11 = c00;
#pragma unroll 4
  for (int k = 0; k < DD; k += 4) {
    v2f a0, a1, b0, b1;
    a0.x = pr0[k]; a0.y = pr0[k + 1];
    a1.x = pr1[k]; a1.y = pr1[k + 1];
    b0.x = qr0[k]; b0.y = qr0[k + 1];
    b1.x = qr1[k]; b1.y = qr1[k + 1];
    c00 = __builtin_amdgcn_wmma_f32_16x16x4_f32(false, a0, false, b0, (short)0, c00, false, false);
    c01 = __builtin_amdgcn_wmma_f32_16x16x4_f32(false, a0, false, b1, (short)0, c01, false, false);
    c10 = __builtin_amdgcn_wmma_f32_16x16x4_f32(false, a1, false, b0, (short)0, c10, false, false);
    c11 = __builtin_amdgcn_wmma_f32_16x16x4_f32(false, a1, false, b1, (short)0, c11, false, false);
  }
  if (mode == 0) {
    const float nq0 = nrm[col0 + idx];
    const float nq1 = nrm[col0 + 16 + idx];
#pragma unroll
    for (int r = 0; r < 8; ++r) {
      const int m = r + half * 8;
      const float np0 = nrm[row0 + m];
      const float np1 = nrm[row0 + 16 + m];
      float* rowp0 = A + (size_t)(row0 + m) * NN + col0 + idx;
      float* rowp1 = A + (size_t)(row0 + 16 + m) * NN + col0 + idx;
      rowp0[0]  = c00[r] * fast_rcp(np0 * nq0 + ZEROF);
      rowp0[16] = c01[r] * fast_rcp(np0 * nq1 + ZEROF);
      rowp1[0]  = c10[r] * fast_rcp(np1 * nq0 + ZEROF);
      rowp1[16] = c11[r] * fast_rcp(np1 * nq1 + ZEROF);
    }
  } else {
#pragma unroll
    for (int r = 0; r < 8; ++r) {
      const int m = r + half * 8;
      float* rowp0 = A + (size_t)(row0 + m) * NN + col0 + idx;
      float* rowp1 = A + (size_t)(row0 + 16 + m) * NN + col0 + idx;
      rowp0[0]  = 0.5f * rowp0[0]  + 0.5f * fast_rcp(1.f + __expf(-c00[r]));
      rowp0[16] = 0.5f * rowp0[16] + 0.5f * fast_rcp(1.f + __expf(-c01[r]));
      rowp1[0]  = 0.5f * rowp1[0]  + 0.5f * fast_rcp(1.f + __expf(-c10[r]));
      rowp1[16] = 0.5f * rowp1[16] + 0.5f * fast_rcp(1.f + __expf(-c11[r]));
    }
  }
}

// ---------------------------------------------------------------------------
// OUT = P * W  (P: NN x 256, W: 256 x 256 rowmajor), K = 256.
// 2x2 tile groups per wave (32x32 output region).
// ---------------------------------------------------------------------------
__global__ void gemm_nn_kernel(const float* __restrict__ P, const float* __restrict__ W,
                               float* __restrict__ OUT) {
  const int wave = threadIdx.x >> 5;
  const int lane = threadIdx.x & 31;
  const int grp  = blockIdx.x * 8 + wave;        // 256*8 groups of 32x32
  const int gi = grp >> 3;
  const int gj = grp & 7;
  const int row0 = gi * 32, col0 = gj * 32;
  const int half = lane >> 4, idx = lane & 15;
  const int kb = half * 2;
  const float* pr0 = P + (size_t)(row0 + idx) * DD + kb;
  const float* pr1 = pr0 + (size_t)16 * DD;
  const float* wc0 = W + (size_t)kb * DD + col0 + idx;
  const float* wc1 = wc0 + 16;
  v8f c00 = {0.f,0.f,0.f,0.f,0.f,0.f,0.f,0.f};
  v8f c01 = c00, c10 = c00, c11 = c00;
#pragma unroll 4
  for (int k = 0; k < DD; k += 4) {
    v2f a0, a1, b0, b1;
    a0.x = pr0[k]; a0.y = pr0[k + 1];
    a1.x = pr1[k]; a1.y = pr1[k + 1];
    b0.x = wc0[(size_t)k * DD]; b0.y = wc0[(size_t)(k + 1) * DD];
    b1.x = wc1[(size_t)k * DD]; b1.y = wc1[(size_t)(k + 1) * DD];
    c00 = __builtin_amdgcn_wmma_f32_16x16x4_f32(false, a0, false, b0, (short)0, c00, false, false);
    c01 = __builtin_amdgcn_wmma_f32_16x16x4_f32(false, a0, false, b1, (short)0, c01, false, false);
    c10 = __builtin_amdgcn_wmma_f32_16x16x4_f32(false, a1, false, b0, (short)0, c10, false, false);
    c11 = __builtin_amdgcn_wmma_f32_16x16x4_f32(false, a1, false, b1, (short)0, c11, false, false);
  }
#pragma unroll
  for (int r = 0; r < 8; ++r) {
    const int m = r + half * 8;
    float* o0 = OUT + (size_t)(row0 + m) * DD + col0 + idx;
    float* o1 = OUT + (size_t)(row0 + 16 + m) * DD + col0 + idx;
    o0[0]  = c00[r];
    o0[16] = c01[r];
    o1[0]  = c10[r];
    o1[16] = c11[r];
  }
}

// ---------------------------------------------------------------------------
// row sums of A (one block per row)
// ---------------------------------------------------------------------------
__global__ void rowsum_kernel(const float* __restrict__ A, float* __restrict__ rs) {
  __shared__ float sh[256];
  const int row = blockIdx.x, t = threadIdx.x;
  const float* p = A + (size_t)row * NN;
  float s = 0.f;
  for (int j = t; j < NN; j += 256) s += p[j];
  sh[t] = s; __syncthreads();
  for (int o = 128; o > 0; o >>= 1) { if (t < o) sh[t] += sh[t + o]; __syncthreads(); }
  if (t == 0) rs[row] = sh[0];
}

// rs[i] -> dinv[i] = rsqrt(|rs[i]| + eps)   (in place)
__global__ void dinv_kernel(float* __restrict__ rs) {
  const int i = blockIdx.x * 256 + threadIdx.x;
  rs[i] = rsqrtf(fabsf(rs[i]) + ZEROF);
}

// ---------------------------------------------------------------------------
// A[i,j] *= dinv[i] * dinv[j]   grid: (NN/1024, NN); 4 columns per thread
// ---------------------------------------------------------------------------
__global__ void dinvscale_kernel(float* __restrict__ A, const float* __restrict__ dinv) {
  const int i = blockIdx.y;
  const int j = (blockIdx.x * 256 + threadIdx.x) * 4;
  const float di = dinv[i];
  float4 r = *(const float4*)(dinv + j);
  float4 v = *(float4*)(A + (size_t)i * NN + j);
  v.x *= di * r.x; v.y *= di * r.y; v.z *= di * r.z; v.w *= di * r.w;
  *(float4*)(A + (size_t)i * NN + j) = v;
}

// ---------------------------------------------------------------------------
// Hnext[n, d] = b0[d] + b1[d] + b2[d]   (bias init, summed across 3 GCN types)
// ---------------------------------------------------------------------------
__global__ void init_hnext_kernel(float* __restrict__ Hn, const float* __restrict__ b3) {
  const int id = blockIdx.x * 256 + threadIdx.x;   // NN*DD threads
  const int d = id & (DD - 1);
  Hn[id] = b3[d] + b3[DD + d] + b3[2 * DD + d];
}

// ---------------------------------------------------------------------------
// edge scatter: Hnext[dst] += HW[src] * w   (one wave per edge, 8 cols/lane)
// useA=0: w = ew[e] ; useA=1: w = A[src, dst]
// ---------------------------------------------------------------------------
__global__ void scatter_kernel(const float* __restrict__ HW,
                               const long long* __restrict__ ei,
                               const float* __restrict__ ew,
                               const float* __restrict__ A,
                               int useA, float* __restrict__ Hn) {
  const int e    = blockIdx.x * 8 + (threadIdx.x >> 5);
  const int lane = threadIdx.x & 31;
  const int src = (int)ei[e];
  const int dst = (int)ei[EE + e];
  const float w = useA ? A[(size_t)src * NN + dst] : ew[e];
  const float* s = HW + (size_t)src * DD + lane * 8;
  float* d       = Hn + (size_t)dst * DD + lane * 8;
  float4 v0 = *(const float4*)(s);
  float4 v1 = *(const float4*)(s + 4);
  atomicAdd(d + 0, v0.x * w); atomicAdd(d + 1, v0.y * w);
  atomicAdd(d + 2, v0.z * w); atomicAdd(d + 3, v0.w * w);
  atomicAdd(d + 4, v1.x * w); atomicAdd(d + 5, v1.y * w);
  atomicAdd(d + 6, v1.z * w); atomicAdd(d + 7, v1.w * w);
}

__global__ void relu_kernel(float* __restrict__ H) {
  const int id = blockIdx.x * 256 + threadIdx.x;
  H[id] = fmaxf(H[id], 0.f);
}

// ---------------------------------------------------------------------------
// finalize: diagonal-block reductions + min/max/sum normalization
// one block per batch b (32 blocks x 256 threads)
// out[0:8192] = map_entity, out[8192:16384] = map_pattern
// ---------------------------------------------------------------------------
__global__ void finalize_kernel(const float* __restrict__ A,
                                const unsigned char* __restrict__ ent,
                                const unsigned char* __restrict__ pat,
                                float* __restrict__ out) {
  __shared__ float sh[256];
  const int b = blockIdx.x, i = threadIdx.x;
  const float* row = A + ((size_t)(b * NPGC + i)) * NN + (size_t)b * NPGC;
  float s = 0.f, se = 0.f, sp = 0.f;
  for (int j = 0; j < NPGC; ++j) {
    const float v = row[j];
    s += v;
    if (ent[b * NPGC + j]) se += v;
    if (pat[b * NPGC + j]) sp += v;
  }
  float me = s - sp;   // map_entity  = A_sum - pat_cols
  float mp = s - se;   // map_pattern = A_sum - ent_cols

#pragma unroll
  for (int which = 0; which < 2; ++which) {
    float m = which ? mp : me;
    sh[i] = m; __syncthreads();
    for (int o = 128; o > 0; o >>= 1) { if (i < o) sh[i] = fminf(sh[i], sh[i + o]); __syncthreads(); }
    const float mn = sh[0]; __syncthreads();
    sh[i] = m; __syncthreads();
    for (int o = 128; o > 0; o >>= 1) { if (i < o) sh[i] = fmaxf(sh[i], sh[i + o]); __syncthreads(); }
    const float mx = sh[0]; __syncthreads();
    m = (m - mn) / (mx - mn + ZEROF);
    sh[i] = m; __syncthreads();
    for (int o = 128; o > 0; o >>= 1) { if (i < o) sh[i] += sh[i + o]; __syncthreads(); }
    const float sm = sh[0]; __syncthreads();
    out[(size_t)which * (BBLK * NPGC) + b * NPGC + i] = m / (sm + ZEROF);
  }
}

// ---------------------------------------------------------------------------
extern "C" void kernel_launch(void* const* d_in, const int* in_sizes, int n_in,
                              void* d_out, int out_size, void* d_ws, size_t ws_size,
                              hipStream_t stream) {
  const float*     x    = (const float*)d_in[0];
  const long long* ei[3] = { (const long long*)d_in[1], (const long long*)d_in[3],
                             (const long long*)d_in[5] };
  const float*     ew[3] = { (const float*)d_in[2], (const float*)d_in[4],
                             (const float*)d_in[6] };
  const unsigned char* ent = (const unsigned char*)d_in[7];
  const unsigned char* pat = (const unsigned char*)d_in[8];
  const float* Wg = (const float*)d_in[9];    // (L,3,D,D)
  const float* bg = (const float*)d_in[10];   // (L,3,D)
  const float* Td = (const float*)d_in[11];   // (L,D,D)
  float* out = (float*)d_out;

  // workspace layout
  float* A   = (float*)d_ws;                       // NN*NN
  float* H   = A   + (size_t)NN * NN;              // NN*DD
  float* Hn  = H   + (size_t)NN * DD;              // NN*DD
  float* HW  = Hn  + (size_t)NN * DD;              // NN*DD
  float* nrm = HW  + (size_t)NN * DD;              // NN
  float* rs  = nrm + NN;                           // NN

  hipMemcpyAsync(H, x, (size_t)NN * DD * sizeof(float),
                 hipMemcpyDeviceToDevice, stream);

  // A = normalized correlation matrix of H
  rownorm_kernel<<<NN / 8, 256, 0, stream>>>(H, nrm);
  gemm_nt_kernel<<<(NN / 32) * (NN / 32) / 8, 256, 0, stream>>>(H, H, nrm, A, 0);
  rowsum_kernel<<<NN, 256, 0, stream>>>(A, rs);
  dinv_kernel<<<NN / 256, 256, 0, stream>>>(rs);
  dinvscale_kernel<<<dim3(NN / 1024, NN), 256, 0, stream>>>(A, rs);

  for (int i = 0; i < LLAY; ++i) {
    init_hnext_kernel<<<(NN * DD) / 256, 256, 0, stream>>>(Hn, bg + (size_t)i * 3 * DD);
    for (int t = 0; t < 3; ++t) {
      gemm_nn_kernel<<<(NN / 32) * (DD / 32) / 8, 256, 0, stream>>>(
          H, Wg + ((size_t)i * 3 + t) * DD * DD, HW);
      scatter_kernel<<<EE / 8, 256, 0, stream>>>(HW, ei[t], ew[t], A, i > 0 ? 1 : 0, Hn);
    }
    relu_kernel<<<(NN * DD) / 256, 256, 0, stream>>>(Hn);
    { float* tmp = H; H = Hn; Hn = tmp; }        // H <- relu(He+Hp+Ho)
    // A = 0.5*A + 0.5*sigmoid((H @ T) @ H^T)
    gemm_nn_kernel<<<(NN / 32) * (DD / 32) / 8, 256, 0, stream>>>(
        H, Td + (size_t)i * DD * DD, HW);
    gemm_nt_kernel<<<(NN / 32) * (NN / 32) / 8, 256, 0, stream>>>(HW, H, nrm, A, 1);
  }

  finalize_kernel<<<BBLK, 256, 0, stream>>>(A, ent, pat, out);
}